// Encoder_15401752723987
// MI455X (gfx1250) — compile-verified
//
#include <hip/hip_runtime.h>

typedef __attribute__((ext_vector_type(2))) float v2f;
typedef __attribute__((ext_vector_type(8))) float v8f;

#define NPTS (1 << 20)

// ---------------------------------------------------------------------------
// SH coefficient matrix B[20][16]:  SH_n = sum_k  mono_k * B[k][n]
// Monomial order k: 1, x, y, z, x2, y2, z2, xy, yz, xz,
//                   x3, y3, z3, x2y, x2z, xy2, y2z, xz2, yz2, xyz
// ---------------------------------------------------------------------------
__constant__ float SH_COEF[20][16] = {
  /* 1   */ {0.28209479177387814f,0,0,0,0,0,-0.31539156525252f,0,0,0,0,0,0,0,0,0},
  /* x   */ {0,0,0,-0.48860251190291987f,0,0,0,0,0,0,0,0,0,0.4570457994644657f,0,0},
  /* y   */ {0,-0.48860251190291987f,0,0,0,0,0,0,0,0,0,0.4570457994644657f,0,0,0,0},
  /* z   */ {0,0,0.48860251190291987f,0,0,0,0,0,0,0,0,0,-1.1195289977703462f,0,0,0},
  /* x2  */ {0,0,0,0,0,0,0,0,0.5462742152960396f,0,0,0,0,0,0,0},
  /* y2  */ {0,0,0,0,0,0,0,0,-0.5462742152960396f,0,0,0,0,0,0,0},
  /* z2  */ {0,0,0,0,0,0,0.94617469575756f,0,0,0,0,0,0,0,0,0},
  /* xy  */ {0,0,0,0,1.0925484305920792f,0,0,0,0,0,0,0,0,0,0,0},
  /* yz  */ {0,0,0,0,0,-1.0925484305920792f,0,0,0,0,0,0,0,0,0,0},
  /* xz  */ {0,0,0,0,0,0,0,-1.0925484305920792f,0,0,0,0,0,0,0,0},
  /* x3  */ {0,0,0,0,0,0,0,0,0,0,0,0,0,0,0,-0.5900435899266435f},
  /* y3  */ {0,0,0,0,0,0,0,0,0,0.5900435899266435f,0,0,0,0,0,0},
  /* z3  */ {0,0,0,0,0,0,0,0,0,0,0,0,1.865881662950577f,0,0,0},
  /* x2y */ {0,0,0,0,0,0,0,0,0,-1.7701307697799304f,0,0,0,0,0,0},
  /* x2z */ {0,0,0,0,0,0,0,0,0,0,0,0,0,0,1.445305721320277f,0},
  /* xy2 */ {0,0,0,0,0,0,0,0,0,0,0,0,0,0,0,1.7701307697799304f},
  /* y2z */ {0,0,0,0,0,0,0,0,0,0,0,0,0,0,-1.445305721320277f,0},
  /* xz2 */ {0,0,0,0,0,0,0,0,0,0,0,0,0,-2.2852289973223288f,0,0},
  /* yz2 */ {0,0,0,0,0,0,0,0,0,0,0,-2.2852289973223288f,0,0,0,0},
  /* xyz */ {0,0,0,0,0,0,0,0,0,0,2.890611442640554f,0,0,0,0,0},
};

// ---------------------------------------------------------------------------
// Kernel 1: trilinear interp of 8 dense grids (L2-resident gathers).
// One thread per point; float2 (b64) corner loads; writes out[p*32 + 2l + f].
// ---------------------------------------------------------------------------
__global__ __launch_bounds__(256) void interp_kernel(
    const float* __restrict__ pos,
    const float* __restrict__ g0, const float* __restrict__ g1,
    const float* __restrict__ g2, const float* __restrict__ g3,
    const float* __restrict__ g4, const float* __restrict__ g5,
    const float* __restrict__ g6, const float* __restrict__ g7,
    float* __restrict__ out)
{
  const int p = blockIdx.x * 256 + threadIdx.x;
  const float px = pos[3 * p + 0];
  const float py = pos[3 * p + 1];
  const float pz = pos[3 * p + 2];

  const float* grids[8] = {g0, g1, g2, g3, g4, g5, g6, g7};
  const int RES[8] = {16, 21, 28, 39, 52, 70, 95, 128};

  float* o = out + (size_t)p * 32;

#pragma unroll
  for (int l = 0; l < 8; ++l) {
    const int R = RES[l];
    const float cx = px * (float)(R - 1);
    const float cy = py * (float)(R - 1);
    const float cz = pz * (float)(R - 1);
    int ix = (int)cx; ix = ix < R - 2 ? ix : R - 2; ix = ix > 0 ? ix : 0;
    int iy = (int)cy; iy = iy < R - 2 ? iy : R - 2; iy = iy > 0 ? iy : 0;
    int iz = (int)cz; iz = iz < R - 2 ? iz : R - 2; iz = iz > 0 ? iz : 0;
    const float fx = cx - (float)ix;
    const float fy = cy - (float)iy;
    const float fz = cz - (float)iz;

    const v2f* __restrict__ g = (const v2f*)grids[l];
    const int i000 = (ix * R + iy) * R + iz;
    const int i100 = i000 + R * R;

    const v2f c000 = g[i000],         c001 = g[i000 + 1];
    const v2f c010 = g[i000 + R],     c011 = g[i000 + R + 1];
    const v2f c100 = g[i100],         c101 = g[i100 + 1];
    const v2f c110 = g[i100 + R],     c111 = g[i100 + R + 1];

    const v2f c00 = c000 + fz * (c001 - c000);
    const v2f c01 = c010 + fz * (c011 - c010);
    const v2f c10 = c100 + fz * (c101 - c100);
    const v2f c11 = c110 + fz * (c111 - c110);
    const v2f c0  = c00 + fy * (c01 - c00);
    const v2f c1  = c10 + fy * (c11 - c10);
    const v2f r   = c0  + fx * (c1 - c0);

    o[2 * l + 0] = r.x;
    o[2 * l + 1] = r.y;
  }
}

// ---------------------------------------------------------------------------
// Kernel 2: SH(l<=3, 16 coeffs) as tall-skinny GEMM via V_WMMA_F32_16X16X4_F32.
// Each wave32 handles a 16-point tile: A = 16x20 monomials (5 chunks of K=4),
// B = constant 20x16 coefficient matrix, D = 16x16 SH block.
// A layout: lanes 0-15 hold M=0..15 / K={c,c+1}; lanes 16-31 same M, K={c+2,c+3}.
// B layout: VGPR0 = K=c (lo lanes) / K=c+2 (hi lanes), VGPR1 = K=c+1 / K=c+3.
// D layout: VGPR r -> M=r (lo) / M=r+8 (hi), N = lane&15.
// ---------------------------------------------------------------------------
__global__ __launch_bounds__(256) void sh_wmma_kernel(
    const float* __restrict__ dir, float* __restrict__ out)
{
  const int lane = threadIdx.x & 31;
  const int wave = threadIdx.x >> 5;
  const int tile = blockIdx.x * 8 + wave;      // 16 points per tile
  const int  n   = lane & 15;                  // column (point within tile / SH idx)
  const bool hi  = lane >= 16;

  const int p = tile * 16 + n;                 // each lane computes monomials for point (lane&15)
  const float dx = dir[3 * p + 0];
  const float dy = dir[3 * p + 1];
  const float dz = dir[3 * p + 2];
  const float inv = rsqrtf(dx * dx + dy * dy + dz * dz);
  const float x = dx * inv, y = dy * inv, z = dz * inv;
  const float xx = x * x, yy = y * y, zz = z * z;

  const float m0 = 1.0f, m1 = x, m2 = y, m3 = z;
  const float m4 = xx, m5 = yy, m6 = zz, m7 = x * y, m8 = y * z, m9 = x * z;
  const float m10 = xx * x, m11 = yy * y, m12 = zz * z;
  const float m13 = xx * y, m14 = xx * z, m15 = x * yy;
  const float m16 = yy * z, m17 = x * zz, m18 = y * zz, m19 = x * y * z;

  v8f acc = {0.f, 0.f, 0.f, 0.f, 0.f, 0.f, 0.f, 0.f};

#define SH_CHUNK(C, E0, E1, E2, E3)                                           \
  do {                                                                        \
    v2f a, b;                                                                 \
    a.x = hi ? (E2) : (E0);                                                   \
    a.y = hi ? (E3) : (E1);                                                   \
    const int kb = (C) + (hi ? 2 : 0);                                        \
    b.x = SH_COEF[kb][n];                                                     \
    b.y = SH_COEF[kb + 1][n];                                                 \
    acc = __builtin_amdgcn_wmma_f32_16x16x4_f32(                              \
        false, a, false, b, (short)0, acc, false, false);                     \
  } while (0)

  SH_CHUNK(0,  m0,  m1,  m2,  m3);
  SH_CHUNK(4,  m4,  m5,  m6,  m7);
  SH_CHUNK(8,  m8,  m9,  m10, m11);
  SH_CHUNK(12, m12, m13, m14, m15);
  SH_CHUNK(16, m16, m17, m18, m19);
#undef SH_CHUNK

  // Scatter D: VGPR r -> row (tile*16 + r [+8 for hi lanes]), SH index n.
  const int rowBase = tile * 16 + (hi ? 8 : 0);
#pragma unroll
  for (int r = 0; r < 8; ++r) {
    out[(size_t)(rowBase + r) * 32 + 16 + n] = acc[r];
  }
}

// ---------------------------------------------------------------------------
extern "C" void kernel_launch(void* const* d_in, const int* in_sizes, int n_in,
                              void* d_out, int out_size, void* d_ws, size_t ws_size,
                              hipStream_t stream) {
  (void)in_sizes; (void)n_in; (void)out_size; (void)d_ws; (void)ws_size;

  const float* pos = (const float*)d_in[0];
  const float* dir = (const float*)d_in[1];
  const float* g0 = (const float*)d_in[2];
  const float* g1 = (const float*)d_in[3];
  const float* g2 = (const float*)d_in[4];
  const float* g3 = (const float*)d_in[5];
  const float* g4 = (const float*)d_in[6];
  const float* g5 = (const float*)d_in[7];
  const float* g6 = (const float*)d_in[8];
  const float* g7 = (const float*)d_in[9];
  float* out = (float*)d_out;

  // Feature interpolation: 1 thread / point, 4096 blocks of 256.
  interp_kernel<<<NPTS / 256, 256, 0, stream>>>(pos, g0, g1, g2, g3, g4, g5,
                                                g6, g7, out);

  // SH via WMMA: 65536 tiles of 16 points, 8 waves (tiles) per 256-thread block.
  sh_wmma_kernel<<<(NPTS / 16) / 8, 256, 0, stream>>>(dir, out);
}